// AdaptiveFeaturePooling_30983894073584
// MI455X (gfx1250) — compile-verified
//
#include <hip/hip_runtime.h>
#include <hip/hip_bf16.h>
#include <math.h>

// Problem constants (from reference setup_inputs)
#define FB   2          // batch
#define FC   256        // channels
#define FH   200        // feature H
#define FW   304        // feature W
#define FHW  (FH * FW)  // 60800
#define NROI 512
#define POOLED 7
#define NBINS (POOLED * POOLED)  // 49
#define SCALE 0.25f

typedef unsigned int u32x4 __attribute__((ext_vector_type(4)));
typedef int          i32x4 __attribute__((ext_vector_type(4)));
typedef int          i32x8 __attribute__((ext_vector_type(8)));

// -------------------------------------------------------------------------
// Kernel 1 (primary): NCHW -> NHWC transpose using the Tensor Data Mover.
// One TDM descriptor per block fetches a 32(C) x 128(P) tile into LDS with
// hardware row padding (128 -> 129 floats) for conflict-free transposed
// reads. 256 threads/block then drain it as coalesced NHWC b128 stores.
// C=256 % 32 == 0 and HW=60800 % 128 == 0 -> no edge cases.
// -------------------------------------------------------------------------
__global__ void nchw_to_nhwc_tdm_kernel(const float* __restrict__ in,
                                        float* __restrict__ out) {
    __shared__ float tile[32][129];  // row stride 129: TDM pad_amount=1 DWORD

    const int t = threadIdx.x;               // 0..255
    const int pTiles = FHW / 128;            // 475
    const int cTiles = FC / 32;              // 8

    int blk = blockIdx.x;
    int b   = blk / (pTiles * cTiles);
    int r   = blk % (pTiles * cTiles);
    int ct  = r / pTiles;
    int pt  = r % pTiles;
    int c0  = ct * 32;
    int p0  = pt * 128;

    const float* gtile = in + ((size_t)(b * FC + c0) * FHW + (size_t)p0);

#if defined(__HIP_DEVICE_COMPILE__)
    if (t < 32) {  // wave 0 only: TDM ignores EXEC; avoid duplicate DMAs
        unsigned long long gaddr = (unsigned long long)(size_t)gtile;  // byte addr
        unsigned int ldsaddr = (unsigned int)(size_t)(&tile[0][0]);    // LDS byte off

        // ---- D# group 0: count=1 | lds_addr | global_addr[56:0] | type=2
        u32x4 g0;
        g0.x = 1u;                                   // count=1 (valid), user mode
        g0.y = ldsaddr;                              // lds_addr
        g0.z = (unsigned int)gaddr;                  // global_addr[31:0]
        g0.w = (unsigned int)((gaddr >> 32) & 0x1FFFFFFu) | (2u << 30); // [56:32]|type=2

        // ---- D# group 1
        // dword0: data_size=2 (4B) | pad_enable | pad_interval=6 (128 DW) |
        //         pad_amount=0 (1 DW)  -> LDS rows of 129 floats
        i32x8 g1;
        g1[0] = (int)((2u << 16) | (1u << 20) | (6u << 22));
        g1[1] = (int)(128u << 16);      // tensor_dim0[15:0]=128 in bits[31:16]
        g1[2] = (int)(32u << 16);       // tensor_dim0[31:16]=0 | tensor_dim1[15:0]=32
        g1[3] = (int)(128u << 16);      // tensor_dim1[31:16]=0 | tile_dim0=128
        g1[4] = (int)(32u);             // tile_dim1=32 | tile_dim2=0
        g1[5] = (int)(FHW);             // tensor_dim0_stride[31:0] = 60800
        g1[6] = 0;                      // stride0[47:32]=0 | stride1[15:0]=0
        g1[7] = 0;                      // stride1[47:16]=0

        i32x4 g2 = {0, 0, 0, 0};        // 2-D tile: dims 2..4 unused
        i32x4 g3 = {0, 0, 0, 0};
        i32x8 g4 = {0, 0, 0, 0, 0, 0, 0, 0};  // 6-arg toolchain: extra group, zeros

        __builtin_amdgcn_tensor_load_to_lds(g0, g1, g2, g3, g4, /*cpol=*/0);
        __builtin_amdgcn_s_wait_tensorcnt(0);
    }
#else
    if (t < 32) {  // host-pass placeholder (never executed)
        for (int row = 0; row < 32; ++row)
            for (int col = t; col < 128; col += 32)
                tile[row][col] = gtile[(size_t)row * FHW + col];
    }
#endif
    __syncthreads();

    // ---- drain: thread t writes float4 of channels at (p0+pi, c0+ci)
    const int ci = (t & 7) << 2;       // 0,4,...,28
    #pragma unroll
    for (int k = 0; k < 4; ++k) {
        const int pi = (t >> 3) + 32 * k;   // 0..127
        float4 v;
        v.x = tile[ci + 0][pi];
        v.y = tile[ci + 1][pi];
        v.z = tile[ci + 2][pi];
        v.w = tile[ci + 3][pi];
        float* gdst = out + ((size_t)b * FHW + (size_t)(p0 + pi)) * FC + (size_t)(c0 + ci);
        *reinterpret_cast<float4*>(gdst) = v;
    }
}

// -------------------------------------------------------------------------
// Kernel 1b (kept in bundle; exercises the async global->LDS path):
// 32x32 tile transpose with per-lane global_load_async_to_lds_b128.
// -------------------------------------------------------------------------
__global__ void nchw_to_nhwc_async_kernel(const float* __restrict__ in,
                                          float* __restrict__ out) {
    __shared__ float tile[32][33];

    const int t = threadIdx.x;                 // 0..255
    const int pTiles = FHW / 32;               // 1900
    const int cTiles = FC / 32;                // 8

    int blk = blockIdx.x;
    int b   = blk / (pTiles * cTiles);
    int r   = blk % (pTiles * cTiles);
    int ct  = r / pTiles;
    int pt  = r % pTiles;
    int c0  = ct * 32;
    int p0  = pt * 32;

    int i = t >> 3;
    int j = (t & 7) << 2;
    const float* gsrc = in + ((size_t)(b * FC + c0 + i) * FHW + (size_t)(p0 + j));

#if defined(__HIP_DEVICE_COMPILE__)
    {
        unsigned int ldsoff = (unsigned int)(size_t)(&tile[i][j]);
        asm volatile("global_load_async_to_lds_b128 %0, %1, off"
                     :: "v"(ldsoff), "v"(gsrc)
                     : "memory");
        asm volatile("s_wait_asynccnt 0x0" ::: "memory");
    }
#else
    {
        tile[i][j + 0] = gsrc[0];
        tile[i][j + 1] = gsrc[1];
        tile[i][j + 2] = gsrc[2];
        tile[i][j + 3] = gsrc[3];
    }
#endif
    __syncthreads();

    int pi = t >> 3;
    int ci = (t & 7) << 2;
    float4 v;
    v.x = tile[ci + 0][pi];
    v.y = tile[ci + 1][pi];
    v.z = tile[ci + 2][pi];
    v.w = tile[ci + 3][pi];
    float* gdst = out + ((size_t)b * FHW + (size_t)(p0 + pi)) * FC + (size_t)(c0 + ci);
    *reinterpret_cast<float4*>(gdst) = v;
}

// -------------------------------------------------------------------------
// Kernel 2: ROI Align (max) on the NHWC staging buffer.
// 64 threads per (roi, ph, pw) bin, 4 channels per thread (float4).
// Corner gathers are 512B-contiguous per wave -> fully coalesced b128.
// -------------------------------------------------------------------------
__global__ void roi_align_nhwc_kernel(const float* __restrict__ feat_nhwc,
                                      const float* __restrict__ boxes,
                                      const int*   __restrict__ batch_idx,
                                      float* __restrict__ out) {
    const int t     = threadIdx.x;     // 0..255
    const int group = t >> 6;          // 0..3   (bin within block)
    const int lane  = t & 63;          // channel group
    const int bin   = blockIdx.x * 4 + group;    // 0 .. NROI*NBINS-1

    const int n  = bin / NBINS;
    const int pp = bin % NBINS;
    const int ph = pp / POOLED;
    const int pw = pp % POOLED;
    const int c  = lane << 2;          // 0..252

    const float x1 = boxes[n * 4 + 0] * SCALE;
    const float y1 = boxes[n * 4 + 1] * SCALE;
    const float x2 = boxes[n * 4 + 2] * SCALE;
    const float y2 = boxes[n * 4 + 3] * SCALE;
    const float roi_w = fmaxf(x2 - x1, 1.0f);
    const float roi_h = fmaxf(y2 - y1, 1.0f);
    const float bszx = roi_w / (float)POOLED;
    const float bszy = roi_h / (float)POOLED;

    const int b = batch_idx[n];
    const float* fb = feat_nhwc + (size_t)b * FHW * FC;

    float4 acc;
    acc.x = acc.y = acc.z = acc.w = -__builtin_inff();

    #pragma unroll
    for (int iy = 0; iy < 2; ++iy) {
        const float sy = ((float)iy + 0.5f) / 2.0f;
        float y = y1 + ((float)ph + sy) * bszy;
        const bool vy = (y > -1.0f) && (y < (float)FH);
        y = fminf(fmaxf(y, 0.0f), (float)(FH - 1));
        const int   y0  = (int)floorf(y);
        const int   yhi = min(y0 + 1, FH - 1);
        const float ly  = y - (float)y0;
        const float hy  = 1.0f - ly;

        #pragma unroll
        for (int ix = 0; ix < 2; ++ix) {
            const float sx = ((float)ix + 0.5f) / 2.0f;
            float x = x1 + ((float)pw + sx) * bszx;
            const bool vx = (x > -1.0f) && (x < (float)FW);
            x = fminf(fmaxf(x, 0.0f), (float)(FW - 1));
            const int   x0  = (int)floorf(x);
            const int   xhi = min(x0 + 1, FW - 1);
            const float lx  = x - (float)x0;
            const float hx  = 1.0f - lx;

            const float w00 = hy * hx, w01 = hy * lx;
            const float w10 = ly * hx, w11 = ly * lx;

            const float4 v00 = *reinterpret_cast<const float4*>(fb + ((size_t)(y0  * FW + x0 )) * FC + c);
            const float4 v01 = *reinterpret_cast<const float4*>(fb + ((size_t)(y0  * FW + xhi)) * FC + c);
            const float4 v10 = *reinterpret_cast<const float4*>(fb + ((size_t)(yhi * FW + x0 )) * FC + c);
            const float4 v11 = *reinterpret_cast<const float4*>(fb + ((size_t)(yhi * FW + xhi)) * FC + c);

            float4 val;
            val.x = w00 * v00.x + w01 * v01.x + w10 * v10.x + w11 * v11.x;
            val.y = w00 * v00.y + w01 * v01.y + w10 * v10.y + w11 * v11.y;
            val.z = w00 * v00.z + w01 * v01.z + w10 * v10.z + w11 * v11.z;
            val.w = w00 * v00.w + w01 * v01.w + w10 * v10.w + w11 * v11.w;

            if (!(vy && vx)) { val.x = 0.0f; val.y = 0.0f; val.z = 0.0f; val.w = 0.0f; }

            acc.x = fmaxf(acc.x, val.x);
            acc.y = fmaxf(acc.y, val.y);
            acc.z = fmaxf(acc.z, val.z);
            acc.w = fmaxf(acc.w, val.w);
        }
    }

    // out is NCHW: (N, C, 7, 7); consecutive channels are 49 floats apart.
    float* o = out + ((size_t)n * FC + (size_t)c) * NBINS + pp;
    o[0 * NBINS] = acc.x;
    o[1 * NBINS] = acc.y;
    o[2 * NBINS] = acc.z;
    o[3 * NBINS] = acc.w;
}

// -------------------------------------------------------------------------
// Fallback: direct gather from NCHW (used only if workspace is too small).
// -------------------------------------------------------------------------
__global__ void roi_align_nchw_kernel(const float* __restrict__ feat,
                                      const float* __restrict__ boxes,
                                      const int*   __restrict__ batch_idx,
                                      float* __restrict__ out,
                                      int total) {
    int idx = blockIdx.x * blockDim.x + threadIdx.x;
    if (idx >= total) return;
    const int pp = idx % NBINS;
    const int c  = (idx / NBINS) % FC;
    const int n  = idx / (NBINS * FC);
    const int ph = pp / POOLED;
    const int pw = pp % POOLED;

    const float x1 = boxes[n * 4 + 0] * SCALE;
    const float y1 = boxes[n * 4 + 1] * SCALE;
    const float x2 = boxes[n * 4 + 2] * SCALE;
    const float y2 = boxes[n * 4 + 3] * SCALE;
    const float roi_w = fmaxf(x2 - x1, 1.0f);
    const float roi_h = fmaxf(y2 - y1, 1.0f);
    const float bszx = roi_w / (float)POOLED;
    const float bszy = roi_h / (float)POOLED;

    const int b = batch_idx[n];
    const float* fc = feat + ((size_t)b * FC + c) * FHW;

    float acc = -__builtin_inff();
    #pragma unroll
    for (int iy = 0; iy < 2; ++iy) {
        float y = y1 + ((float)ph + ((float)iy + 0.5f) / 2.0f) * bszy;
        const bool vy = (y > -1.0f) && (y < (float)FH);
        y = fminf(fmaxf(y, 0.0f), (float)(FH - 1));
        const int   y0  = (int)floorf(y);
        const int   yhi = min(y0 + 1, FH - 1);
        const float ly  = y - (float)y0;
        const float hy  = 1.0f - ly;
        #pragma unroll
        for (int ix = 0; ix < 2; ++ix) {
            float x = x1 + ((float)pw + ((float)ix + 0.5f) / 2.0f) * bszx;
            const bool vx = (x > -1.0f) && (x < (float)FW);
            x = fminf(fmaxf(x, 0.0f), (float)(FW - 1));
            const int   x0  = (int)floorf(x);
            const int   xhi = min(x0 + 1, FW - 1);
            const float lx  = x - (float)x0;
            const float hx  = 1.0f - lx;
            float v = (hy * hx) * fc[y0  * FW + x0 ] + (hy * lx) * fc[y0  * FW + xhi]
                    + (ly * hx) * fc[yhi * FW + x0 ] + (ly * lx) * fc[yhi * FW + xhi];
            if (!(vy && vx)) v = 0.0f;
            acc = fmaxf(acc, v);
        }
    }
    out[idx] = acc;
}

extern "C" void kernel_launch(void* const* d_in, const int* in_sizes, int n_in,
                              void* d_out, int out_size, void* d_ws, size_t ws_size,
                              hipStream_t stream) {
    const float* feature   = (const float*)d_in[0];   // (2,256,200,304) f32
    const float* boxes     = (const float*)d_in[1];   // (512,4) f32
    const int*   batch_idx = (const int*)d_in[2];     // (512,) i32
    float*       out       = (float*)d_out;           // (512,256,7,7) f32

    const size_t nhwc_bytes = (size_t)FB * FHW * FC * sizeof(float);  // ~124.5 MB

    if (ws_size >= nhwc_bytes) {
        float* feat_nhwc = (float*)d_ws;
        const int tGrid = FB * (FC / 32) * (FHW / 128);         // 7600 blocks
        nchw_to_nhwc_tdm_kernel<<<tGrid, 256, 0, stream>>>(feature, feat_nhwc);
        const int rGrid = (NROI * NBINS) / 4;                   // 6272 blocks
        roi_align_nhwc_kernel<<<rGrid, 256, 0, stream>>>(feat_nhwc, boxes, batch_idx, out);
    } else {
        const int total = NROI * FC * NBINS;                    // 6,422,528
        roi_align_nchw_kernel<<<(total + 255) / 256, 256, 0, stream>>>(
            feature, boxes, batch_idx, out, total);
    }
}